// GCN_75634374083120
// MI455X (gfx1250) — compile-verified
//
#include <hip/hip_runtime.h>
#include <hip/hip_bf16.h>

typedef float v2f __attribute__((ext_vector_type(2)));
typedef float v8f __attribute__((ext_vector_type(8)));

#define FDIM 128

// ---------------------------------------------------------------------------
// Degree / normalization kernels
// ---------------------------------------------------------------------------
__global__ void gcn_init_deg(float* __restrict__ deg, int n) {
    int i = blockIdx.x * blockDim.x + threadIdx.x;
    if (i < n) deg[i] = 1.0f;  // self-loop contributes 1 to every node's degree
}

__global__ void gcn_edge_deg(const int* __restrict__ ei, float* __restrict__ deg, int e) {
    int i = blockIdx.x * blockDim.x + threadIdx.x;
    if (i < e) atomicAdd(&deg[ei[e + i]], 1.0f);  // dst row of edge_index
}

__global__ void gcn_deg_to_dinv(float* __restrict__ deg, int n) {
    int i = blockIdx.x * blockDim.x + threadIdx.x;
    if (i < n) {
        float d = deg[i];
        deg[i] = (d > 0.0f) ? rsqrtf(d) : 0.0f;
    }
}

__global__ void gcn_zero(float* __restrict__ p, size_t n) {
    size_t i = (size_t)blockIdx.x * blockDim.x + threadIdx.x;
    size_t stride = (size_t)gridDim.x * blockDim.x;
    for (; i < n; i += stride) p[i] = 0.0f;
}

// ---------------------------------------------------------------------------
// GEMM: C[M x 128] = A[M x 128] * W[128 x 128]  via V_WMMA_F32_16X16X4_F32
// Block = 128 threads (4 waves). Each wave computes a 16-row x 128-col panel
// as 8 accumulator tiles (v8f each). W staged TRANSPOSED in LDS (64 KB) so
// each B fragment (two consecutive K values of one column) is a single
// contiguous ds_load_b64 into an even-aligned VGPR pair.
// M is a multiple of 16 here, so tile validity is wave-granular.
// ---------------------------------------------------------------------------
__global__ void __launch_bounds__(128)
gcn_gemm_wmma(const float* __restrict__ A, const float* __restrict__ W,
              float* __restrict__ C, int M) {
    __shared__ float sWt[FDIM * FDIM];  // 64 KB, transposed: sWt[col*128 + k] = W[k][col]

    const int tid = threadIdx.x;
    // Cooperative transposed staging: coalesced float4 reads from global,
    // scalar DS stores into column-major layout. 32 float4 per thread.
    {
        const float4* Wv = (const float4*)W;
        for (int i = tid; i < (FDIM * FDIM) / 4; i += 128) {
            const int r  = i >> 5;          // source row (K index), 32 float4 per row
            const int c4 = (i & 31) << 2;   // source col base
            const float4 w = Wv[i];
            sWt[(c4 + 0) * FDIM + r] = w.x;
            sWt[(c4 + 1) * FDIM + r] = w.y;
            sWt[(c4 + 2) * FDIM + r] = w.z;
            sWt[(c4 + 3) * FDIM + r] = w.w;
        }
    }
    __syncthreads();

    const int wave  = tid >> 5;
    const int lane  = tid & 31;
    const int m0    = (blockIdx.x * 4 + wave) * 16;
    if (m0 >= M) return;

    const int mrow  = lane & 15;     // A-frag row / B-frag col / D-frag col within tile
    const int khalf = lane >> 4;     // lanes 16-31 hold K=2,3 of each 4-step

    v8f acc[8];
    #pragma unroll
    for (int t = 0; t < 8; ++t) acc[t] = (v8f){0.f, 0.f, 0.f, 0.f, 0.f, 0.f, 0.f, 0.f};

    const float* Arow = A + (size_t)(m0 + mrow) * FDIM;

    for (int k = 0; k < FDIM; k += 4) {
        const int kb = k + 2 * khalf;
        // A fragment: A[m0+mrow][kb], A[m0+mrow][kb+1]  (one global_load_b64)
        v2f a;
        {
            const float2 av = *(const float2*)(Arow + kb);
            a.x = av.x; a.y = av.y;
        }
        #pragma unroll
        for (int t = 0; t < 8; ++t) {
            const int col = t * 16 + mrow;
            // B fragment: W[kb][col], W[kb+1][col] -> contiguous in sWt (ds_load_b64)
            const v2f b = *(const v2f*)(&sWt[col * FDIM + kb]);
            acc[t] = __builtin_amdgcn_wmma_f32_16x16x4_f32(
                false, a, false, b, (short)0, acc[t], false, false);
        }
    }

    // Store: VGPR r of tile t -> row m0 + r + 8*khalf, col t*16 + mrow
    #pragma unroll
    for (int t = 0; t < 8; ++t) {
        float* Cout = C + (size_t)(m0 + 8 * khalf) * FDIM + t * 16 + mrow;
        #pragma unroll
        for (int r = 0; r < 8; ++r) {
            Cout[(size_t)r * FDIM] = acc[t][r];
        }
    }
}

// ---------------------------------------------------------------------------
// Edge scatter: one wave32 per edge; each lane handles 4 contiguous floats.
// agg[dst] += h[src] * dinv[src]*dinv[dst]
// ---------------------------------------------------------------------------
__global__ void gcn_scatter(const int* __restrict__ ei, const float* __restrict__ h,
                            const float* __restrict__ dinv, float* __restrict__ agg,
                            int e) {
    const int edge = (int)(((size_t)blockIdx.x * blockDim.x + threadIdx.x) >> 5);
    const int lane = threadIdx.x & 31;
    if (edge >= e) return;
    const int src = ei[edge];
    const int dst = ei[e + edge];
    const float nrm = dinv[src] * dinv[dst];
    const float4 v = *(const float4*)(h + (size_t)src * FDIM + lane * 4);
    float* o = agg + (size_t)dst * FDIM + lane * 4;
    atomicAdd(o + 0, v.x * nrm);
    atomicAdd(o + 1, v.y * nrm);
    atomicAdd(o + 2, v.z * nrm);
    atomicAdd(o + 3, v.w * nrm);
}

// ---------------------------------------------------------------------------
// Fused self-loop + bias + ReLU:
// out[i][c] = relu(agg[i][c] + h[i][c]*dinv[i]^2 + b[c])   (in-place safe)
// ---------------------------------------------------------------------------
__global__ void gcn_finalize(const float* __restrict__ agg, const float* __restrict__ h,
                             const float* __restrict__ dinv, const float* __restrict__ b,
                             float* __restrict__ out, int n) {
    const int i = blockIdx.x;
    const int c = threadIdx.x;
    if (i >= n) return;
    const float d = dinv[i];
    const size_t idx = (size_t)i * FDIM + c;
    const float v = agg[idx] + h[idx] * d * d + b[c];
    out[idx] = fmaxf(v, 0.0f);
}

// ---------------------------------------------------------------------------
extern "C" void kernel_launch(void* const* d_in, const int* in_sizes, int n_in,
                              void* d_out, int out_size, void* d_ws, size_t ws_size,
                              hipStream_t stream) {
    const float* x  = (const float*)d_in[0];
    const int*   ei = (const int*)  d_in[1];
    const float* W1 = (const float*)d_in[2];
    const float* b1 = (const float*)d_in[3];
    const float* W2 = (const float*)d_in[4];
    const float* b2 = (const float*)d_in[5];
    float* out = (float*)d_out;

    const int n = in_sizes[0] / FDIM;   // 50000 nodes
    const int e = in_sizes[1] / 2;      // 600000 edges

    // Workspace layout (floats): [dinv : nAl][h : n*128][agg : n*128]
    float* ws = (float*)d_ws;
    const size_t nAl = ((size_t)n + 63) & ~(size_t)63;
    float* dinv = ws;
    float* h    = ws + nAl;
    float* agg  = h + (size_t)n * FDIM;
    const size_t feat = (size_t)n * FDIM;

    const int tpb = 256;

    // --- normalization: deg -> dinv (in place) ---
    gcn_init_deg <<<(n + tpb - 1) / tpb, tpb, 0, stream>>>(dinv, n);
    gcn_edge_deg <<<(e + tpb - 1) / tpb, tpb, 0, stream>>>(ei, dinv, e);
    gcn_deg_to_dinv<<<(n + tpb - 1) / tpb, tpb, 0, stream>>>(dinv, n);

    const int gemm_blocks = (n + 63) / 64;  // 4 waves x 16 rows per block

    // --- layer 1 ---
    gcn_gemm_wmma<<<gemm_blocks, 128, 0, stream>>>(x, W1, h, n);
    gcn_zero     <<<2048, tpb, 0, stream>>>(agg, feat);
    gcn_scatter  <<<(e * 32 + tpb - 1) / tpb, tpb, 0, stream>>>(ei, h, dinv, agg, e);
    gcn_finalize <<<n, FDIM, 0, stream>>>(agg, h, dinv, b1, agg, n);  // y1 in agg

    // --- layer 2 ---
    gcn_gemm_wmma<<<gemm_blocks, 128, 0, stream>>>(agg, W2, h, n);
    gcn_zero     <<<2048, tpb, 0, stream>>>(agg, feat);
    gcn_scatter  <<<(e * 32 + tpb - 1) / tpb, tpb, 0, stream>>>(ei, h, dinv, agg, e);
    gcn_finalize <<<n, FDIM, 0, stream>>>(agg, h, dinv, b2, out, n);
}